// DinoGazeSpade_91250875171102
// MI455X (gfx1250) — compile-verified
//
#include <hip/hip_runtime.h>

#define S_SEG 128
#define BATCH 2
#define CCH   768
#define HP    28
#define WP    28
#define P_PATCH 784   // HP*WP
#define HI 392
#define WI 392

typedef float v2f __attribute__((ext_vector_type(2)));
typedef float v8f __attribute__((ext_vector_type(8)));

// ---------------- Phase 1a: per-segment counts (tiny) ----------------
__global__ void __launch_bounds__(256)
counts_kernel(const int* __restrict__ mask, int* __restrict__ counts) {
    __shared__ int cnt[S_SEG];
    const int b   = blockIdx.x;
    const int tid = threadIdx.x;
    if (tid < S_SEG) cnt[tid] = 0;
    __syncthreads();
    for (int p = tid; p < P_PATCH; p += 256) {
        int hp = p / WP;
        int wp = p - hp * WP;
        int m  = mask[b * HI * WI + (hp * 14) * WI + wp * 14];
        m = min(max(m, 0), S_SEG - 1);
        atomicAdd(&cnt[m], 1);
    }
    __syncthreads();
    if (tid < S_SEG) counts[b * S_SEG + tid] = cnt[tid];
}

// ---------------- Phase 1b: segment sums as one-hot GEMM via WMMA ----------------
// sums(128 x 768) = OneHot(128 x 784) * Feat(784 x 768), per batch, fp32 WMMA.
// grid = BATCH * 48 (one wave per block; each block: all 8 M-tiles x one 16-chan N-tile)
#define LDSF_STRIDE 786   // 784 padded to even stride, 786%64=18 -> conflict-free, 8B aligned
__global__ void __launch_bounds__(32)
seg_gemm_kernel(const float* __restrict__ F,
                const int*   __restrict__ mask,
                float*       __restrict__ sums) {
    __shared__ float ldsF[16 * LDSF_STRIDE];
    __shared__ int   gidLds[P_PATCH];

    const int lane = threadIdx.x;          // 0..31
    const int b    = blockIdx.x / 48;
    const int nt   = blockIdx.x % 48;      // 16-channel tile

    // stage patch -> segment ids
    for (int p = lane; p < P_PATCH; p += 32) {
        int hp = p / WP;
        int wp = p - hp * WP;
        int m  = mask[b * HI * WI + (hp * 14) * WI + wp * 14];
        gidLds[p] = min(max(m, 0), S_SEG - 1);
    }
    // stage 16 x 784 feature tile (coalesced along k)
    const float* Fb = F + ((size_t)b * CCH + nt * 16) * P_PATCH;
    for (int ch = 0; ch < 16; ++ch)
        for (int k = lane; k < P_PATCH; k += 32)
            ldsF[ch * LDSF_STRIDE + k] = Fb[ch * P_PATCH + k];
    __syncthreads();

    v8f acc[8];
    const v8f zero = {0.f, 0.f, 0.f, 0.f, 0.f, 0.f, 0.f, 0.f};
#pragma unroll
    for (int mt = 0; mt < 8; ++mt) acc[mt] = zero;

    const int n  = lane & 15;          // A: row M, B: col N
    const int kk = (lane >> 4) * 2;    // K sub-offset per half-wave (ISA 16x4/4x16 fp32 layout)

    for (int k0 = 0; k0 < P_PATCH; k0 += 4) {
        const int k_a = k0 + kk;
        v2f bv;
        bv.x = ldsF[n * LDSF_STRIDE + k_a];
        bv.y = ldsF[n * LDSF_STRIDE + k_a + 1];
        const int g0 = gidLds[k_a];
        const int g1 = gidLds[k_a + 1];
#pragma unroll
        for (int mt = 0; mt < 8; ++mt) {
            v2f av;
            av.x = (g0 == (mt * 16 + n)) ? 1.0f : 0.0f;
            av.y = (g1 == (mt * 16 + n)) ? 1.0f : 0.0f;
            acc[mt] = __builtin_amdgcn_wmma_f32_16x16x4_f32(
                false, av, false, bv, (short)0, acc[mt], false, false);
        }
    }

    // D layout: VGPR j -> M=j (lanes 0-15) / M=j+8 (lanes 16-31); N = lane&15
    const int col    = nt * 16 + n;
    const int rowAdd = (lane >> 4) * 8;
#pragma unroll
    for (int mt = 0; mt < 8; ++mt) {
#pragma unroll
        for (int j = 0; j < 8; ++j) {
            int seg = mt * 16 + j + rowAdd;
            sums[(b * S_SEG + seg) * CCH + col] = acc[mt][j];
        }
    }
}

// ---------------- Phase 2: paint (bandwidth bound, ~944 MB of stores) ----------------
#define CCHUNK 64          // channels per block
#define NCC    (CCH / CCHUNK)   // 12
#define ROWS   7
#define NRG    (HI / ROWS)      // 56
#define TBL_STRIDE 129     // pad: gather bank = (c+s)%64 -> conflict-free
__global__ void __launch_bounds__(256)
paint_kernel(const int*   __restrict__ mask,
             const float* __restrict__ sums,
             const int*   __restrict__ counts,
             float*       __restrict__ out) {
    __shared__ float tbl[CCHUNK * TBL_STRIDE];   // [c][s], transposed avg table slice
    __shared__ float rcp[S_SEG];
    __shared__ int   msk[ROWS * WI];

    const int tid = threadIdx.x;
    const int bid = blockIdx.x;
    const int rg  = bid % NRG;
    const int t   = bid / NRG;
    const int cc  = t % NCC;
    const int b   = t / NCC;
    const int c0  = cc * CCHUNK;
    const int hi0 = rg * ROWS;

    if (tid < S_SEG) {
        int c_ = counts[b * S_SEG + tid];
        rcp[tid] = (c_ > 0) ? (1.0f / (float)c_) : 0.0f;
    }
    for (int r = 0; r < ROWS; ++r)
        for (int w = tid; w < WI; w += 256) {
            int m = mask[(b * HI + hi0 + r) * WI + w];
            msk[r * WI + w] = min(max(m, 0), S_SEG - 1);
        }
    __syncthreads();

    // load + transpose + average the 64x128 table slice (global reads coalesced on c)
    for (int i = tid; i < CCHUNK * S_SEG; i += 256) {
        int c = i & (CCHUNK - 1);
        int s = i >> 6;
        tbl[c * TBL_STRIDE + s] = sums[(b * S_SEG + s) * CCH + c0 + c] * rcp[s];
    }
    __syncthreads();

    // stream out: for each channel, 7 rows x 392 cols as aligned float4 stores
    for (int c = 0; c < CCHUNK; ++c) {
        const float* trow = &tbl[c * TBL_STRIDE];
        float* orow = out + ((size_t)(b * CCH + c0 + c) * HI + hi0) * WI;
        for (int e = tid; e < 98 * ROWS; e += 256) {
            int r  = e / 98;
            int w4 = e - r * 98;
            const int4 sm = *(const int4*)&msk[r * WI + w4 * 4];
            float4 v;
            v.x = trow[sm.x];
            v.y = trow[sm.y];
            v.z = trow[sm.z];
            v.w = trow[sm.w];
            *(float4*)&orow[r * WI + w4 * 4] = v;
        }
    }
}

extern "C" void kernel_launch(void* const* d_in, const int* in_sizes, int n_in,
                              void* d_out, int out_size, void* d_ws, size_t ws_size,
                              hipStream_t stream) {
    (void)in_sizes; (void)n_in; (void)out_size; (void)ws_size;
    const float* F    = (const float*)d_in[0];
    const int*   mask = (const int*)d_in[1];
    float*       out  = (float*)d_out;

    float* sums   = (float*)d_ws;                                            // 256*768 fp32
    int*   counts = (int*)((char*)d_ws + (size_t)BATCH * S_SEG * CCH * sizeof(float));

    counts_kernel<<<BATCH, 256, 0, stream>>>(mask, counts);
    seg_gemm_kernel<<<BATCH * 48, 32, 0, stream>>>(F, mask, sums);
    paint_kernel<<<BATCH * NCC * NRG, 256, 0, stream>>>(mask, sums, counts, out);
}